// MulticlassNeighborLayer_59519656788346
// MI455X (gfx1250) — compile-verified
//
#include <hip/hip_runtime.h>
#include <hip/hip_bf16.h>

typedef __attribute__((ext_vector_type(2))) float v2f;
typedef __attribute__((ext_vector_type(8))) float v8f;

#define NUM_CELLS   16384
#define NUM_CLASSES 64
#define BATCH       64

#if __has_builtin(__builtin_amdgcn_tanhf)
#define FAST_TANH(x) __builtin_amdgcn_tanhf(x)
#else
#define FAST_TANH(x) tanhf(x)
#endif

// out[b,i,c] = x[b,i-1]*W[i,c,0] + x[b,i]*W[i,c,1] + x[b,i+1]*W[i,c,2] + bias[i,c]
// then 1.7159 * tanh(2/3 * out).  Per cell i this is a (64x3)@(3x64) GEMM,
// done as 16x16 tiles with V_WMMA_F32_16X16X4_F32 (K padded 3 -> 4 with zeros).
__global__ __launch_bounds__(256) void
MulticlassNeighborLayer_59519656788346_kernel(const float* __restrict__ x,     // (B, N)
                                              const float* __restrict__ w,     // (N, C, 3)
                                              const float* __restrict__ bias,  // (N, C)
                                              float* __restrict__ out)         // (B, N, C)
{
    const int lane  = threadIdx.x & 31;
    const int gwave = blockIdx.x * 8 + (threadIdx.x >> 5);
    const int i     = gwave >> 2;          // cell index 0..16383
    const int c0    = (gwave & 3) * 16;    // class tile origin
    const int col   = lane & 15;           // N (class) within tile
    const bool hi   = lane >= 16;

    const int im1 = (i == 0) ? (NUM_CELLS - 1) : (i - 1);
    const int ip1 = (i == NUM_CELLS - 1) ? 0 : (i + 1);

    // B tile (K=4 x N=16 f32, 2 VGPRs):
    //   lanes 0-15 : VGPR0 = W[i,c,0] (K0), VGPR1 = W[i,c,1] (K1)
    //   lanes 16-31: VGPR0 = W[i,c,2] (K2), VGPR1 = 0        (K3)
    const float* wrow = w + (size_t)i * (NUM_CLASSES * 3) + (size_t)(c0 + col) * 3;
    v2f bmat;
    bmat.x = hi ? wrow[2] : wrow[0];
    bmat.y = hi ? 0.0f    : wrow[1];

    // C init: bias[i, c0+col] broadcast down all 16 rows (all 8 acc VGPRs).
    const float cb = bias[(size_t)i * NUM_CLASSES + c0 + col];
    v8f cinit;
#pragma unroll
    for (int v = 0; v < 8; ++v) cinit[v] = cb;

    const int rowSel = hi ? (lane - 16) : lane;               // A-matrix row (batch) in half-wave
    const size_t outRowStride = (size_t)NUM_CELLS * NUM_CLASSES;

#pragma unroll
    for (int b0 = 0; b0 < BATCH; b0 += 16) {
        // A tile (M=16 x K=4 f32, 2 VGPRs):
        //   lanes 0-15 : VGPR0 = x[b, i-1] (K0), VGPR1 = x[b, i] (K1)
        //   lanes 16-31: VGPR0 = x[b, i+1] (K2), VGPR1 = 0      (K3)
        const size_t bA = (size_t)(b0 + rowSel) * NUM_CELLS;
        v2f amat;
        amat.x = hi ? x[bA + ip1] : x[bA + im1];
        amat.y = hi ? 0.0f        : x[bA + i];

        // D = A*B + C   (emits v_wmma_f32_16x16x4_f32)
        v8f d = __builtin_amdgcn_wmma_f32_16x16x4_f32(
            /*neg_a=*/false, amat, /*neg_b=*/false, bmat,
            /*c_mod=*/(short)0, cinit, /*reuse_a=*/false, /*reuse_b=*/false);

        // D layout: lane 0-15 -> row v ; lane 16-31 -> row v+8.
        float* outp = out + (size_t)(b0 + (hi ? 8 : 0)) * outRowStride
                          + (size_t)i * NUM_CLASSES + (c0 + col);
#pragma unroll
        for (int v = 0; v < 8; ++v) {
            float t = d[v] * (2.0f / 3.0f);
            outp[(size_t)v * outRowStride] = 1.7159f * FAST_TANH(t);
        }
    }
}

extern "C" void kernel_launch(void* const* d_in, const int* in_sizes, int n_in,
                              void* d_out, int out_size, void* d_ws, size_t ws_size,
                              hipStream_t stream) {
    (void)in_sizes; (void)n_in; (void)d_ws; (void)ws_size; (void)out_size;
    const float* x    = (const float*)d_in[0];   // (64, 16384, 1)
    const float* wts  = (const float*)d_in[1];   // (16384, 64, 3)
    const float* bias = (const float*)d_in[2];   // (16384, 64)
    float* out        = (float*)d_out;           // (64, 16384, 64)

    // 16384 cells * 4 class-tiles = 65536 waves; 8 waves (256 threads) per block.
    const int blocks = (NUM_CELLS * 4) / 8;      // 8192
    MulticlassNeighborLayer_59519656788346_kernel<<<blocks, 256, 0, stream>>>(x, wts, bias, out);
}